// PredictorLG_75453985457423
// MI455X (gfx1250) — compile-verified
//
#include <hip/hip_runtime.h>
#include <hip/hip_bf16.h>

#define BB   4
#define CDIM 64
#define C4   16
#define HGT  512
#define WID  512
#define HW   (HGT * WID)     // 262144
#define NWIN 4096            // windows per image (64x64 of 8x8)
#define DIM  62

typedef __attribute__((ext_vector_type(2))) float v2f;
typedef __attribute__((ext_vector_type(8))) float v8f;

__device__ __forceinline__ float leaky(float x)    { return x >= 0.f ? x : 0.1f * x; }
__device__ __forceinline__ float sigmoidf(float x) { return 1.f / (1.f + __expf(-x)); }

// ---------------------------------------------------------------------------
// Kernel 1: fused 1x1 conv (f32 WMMA 16x16x4, exact fp32 math) -> LayerNorm ->
//           LeakyReLU + offsets MLP -> d_out, channel-mean xm -> ws,
//           NHWC x_act -> ws, deterministic per-block pooled partials -> ws.
// Each wave: D[16 out-ch x 16 pixels] = w_in[16x64] * x[64 x 16 px] + b_in,
// built from 16 chained V_WMMA_F32_16X16X4_F32 steps (K = 4 each).
// f32 A 16x4 layout: lanes 0-15 hold M with K={0,1} in VGPR0/1; lanes 16-31 K={2,3}.
// ---------------------------------------------------------------------------
__global__ void __launch_bounds__(256) main_kernel(
    const float* __restrict__ input_x,
    const float* __restrict__ w_in,  const float* __restrict__ b_in,
    const float* __restrict__ ln_w,  const float* __restrict__ ln_b,
    const float* __restrict__ w_off1, const float* __restrict__ b_off1,
    const float* __restrict__ w_off2, const float* __restrict__ b_off2,
    float* __restrict__ out_offsets,   // [B,2,H,W]
    float* __restrict__ ws_partial,    // [gridDim.x * 16]
    float* __restrict__ ws_xm,         // [B*HW]
    float* __restrict__ ws_xact)       // [B*HW*16] pixel-major (NHWC)
{
    const int tid  = threadIdx.x;
    const int lane = tid & 31;
    const int wave = tid >> 5;
    const int half = lane >> 4;     // which K-pair / channel-half this lane owns
    const int nIdx = lane & 15;     // A: out-channel row M; B: pixel column N

    const int blocksPerImage = HW / 128;              // 2048
    const int b   = blockIdx.x / blocksPerImage;
    const int bim = blockIdx.x % blocksPerImage;
    const int p   = bim * 128 + wave * 16 + nIdx;     // pixel within image

    const float* xin  = input_x + (size_t)b * CDIM * HW;
    const float* wrow = w_in + nIdx * CDIM;

    // Accumulator seeded with bias: VGPR r holds channel (r + 8*half) of pixel p.
    v8f acc;
    #pragma unroll
    for (int r = 0; r < 8; ++r) acc[r] = b_in[r + 8 * half];

    // 16 WMMA steps over K = 64 (exact fp32, matches reference numerics).
    #pragma unroll
    for (int s = 0; s < 16; ++s) {
        const int k0 = 4 * s + 2 * half;
        v2f af, bf;
        af[0] = wrow[k0];
        af[1] = wrow[k0 + 1];
        bf[0] = xin[(size_t)k0 * HW + p];
        bf[1] = xin[(size_t)(k0 + 1) * HW + p];
        acc = __builtin_amdgcn_wmma_f32_16x16x4_f32(false, af, false, bf,
                                                    (short)0, acc, false, false);
    }

    // Channel LayerNorm over 16 channels: each lane has 8, partner (lane^16) has 8.
    float s1 = 0.f, s2 = 0.f;
    #pragma unroll
    for (int r = 0; r < 8; ++r) { s1 += acc[r]; s2 += acc[r] * acc[r]; }
    s1 += __shfl_xor(s1, 16, 32);
    s2 += __shfl_xor(s2, 16, 32);
    const float mean = s1 * (1.f / 16.f);
    const float var  = s2 * (1.f / 16.f) - mean * mean;
    const float inv  = rsqrtf(var + 1e-6f);

    float y[8];
    float ysum = 0.f;
    #pragma unroll
    for (int r = 0; r < 8; ++r) {
        int ch = r + 8 * half;
        float v = ln_w[ch] * ((acc[r] - mean) * inv) + ln_b[ch];
        v = leaky(v);
        y[r] = v;
        ysum += v;
    }

    // x_act NHWC store (for the 3x3 conv kernel): two float4s per lane.
    {
        float4* dst = (float4*)(ws_xact + ((size_t)b * HW + p) * 16 + 8 * half);
        dst[0] = make_float4(y[0], y[1], y[2], y[3]);
        dst[1] = make_float4(y[4], y[5], y[6], y[7]);
    }

    // xm = mean over 16 channels
    {
        float tot = ysum + __shfl_xor(ysum, 16, 32);
        if (half == 0) ws_xm[(size_t)b * HW + p] = tot * (1.f / 16.f);
    }

    // offsets branch: 16->8 leaky, 8->2 tanh * 8
    float o1[8];
    #pragma unroll
    for (int o = 0; o < 8; ++o) {
        float s = 0.f;
        #pragma unroll
        for (int r = 0; r < 8; ++r) s += w_off1[o * C4 + (r + 8 * half)] * y[r];
        o1[o] = s;
    }
    #pragma unroll
    for (int o = 0; o < 8; ++o)
        o1[o] = leaky(o1[o] + __shfl_xor(o1[o], 16, 32) + b_off1[o]);
    float l0 = b_off2[0], l1 = b_off2[1];
    #pragma unroll
    for (int o = 0; o < 8; ++o) { l0 += w_off2[o] * o1[o]; l1 += w_off2[8 + o] * o1[o]; }
    const float offs = (half == 0) ? tanhf(l0) * 8.f : tanhf(l1) * 8.f;
    out_offsets[((size_t)b * 2 + half) * HW + p] = offs;

    // Deterministic per-block pooled partials (no atomics): LDS stage + fixed-order sum.
    __shared__ float sAcc[256][8];
    #pragma unroll
    for (int r = 0; r < 8; ++r) sAcc[tid][r] = y[r];
    __syncthreads();
    if (tid < 16) {
        int r = tid & 7, hh = tid >> 3;
        float s = 0.f;
        for (int t = 0; t < 256; ++t)
            if (((t >> 4) & 1) == hh) s += sAcc[t][r];
        ws_partial[(size_t)blockIdx.x * 16 + tid] = s;
    }
}

// ---------------------------------------------------------------------------
// Kernel 2: spatial attention — 3x3 conv (16->1, zero 'SAME' pad) + sigmoid.
// Reads the L2-resident NHWC activation.
// ---------------------------------------------------------------------------
__global__ void __launch_bounds__(256) sa_kernel(
    const float* __restrict__ ws_xact,
    const float* __restrict__ w_sa, const float* __restrict__ b_sa,
    float* __restrict__ out_sa)
{
    __shared__ float sW[144];          // [16][3][3]
    const int tid = threadIdx.x;
    if (tid < 144) sW[tid] = w_sa[tid];
    __syncthreads();

    const size_t g = (size_t)blockIdx.x * 256 + tid;
    const int b = (int)(g / HW);
    const int p = (int)(g % HW);
    const int h = p / WID, w = p % WID;

    float acc = b_sa[0];
    #pragma unroll
    for (int kh = -1; kh <= 1; ++kh) {
        int hh = h + kh;
        if (hh < 0 || hh >= HGT) continue;
        #pragma unroll
        for (int kw = -1; kw <= 1; ++kw) {
            int wp = w + kw;
            if (wp < 0 || wp >= WID) continue;
            const float* px = ws_xact + ((size_t)b * HW + (size_t)hh * WID + wp) * 16;
            const int widx = (kh + 1) * 3 + (kw + 1);
            #pragma unroll
            for (int c = 0; c < 16; ++c) acc += px[c] * sW[c * 9 + widx];
        }
    }
    out_sa[g] = sigmoidf(acc);
}

// ---------------------------------------------------------------------------
// Kernel 3: channel attention — reduce partials (fixed order) -> mean -> 16->62 FC -> sigmoid
// ---------------------------------------------------------------------------
__global__ void __launch_bounds__(256) ca_kernel(
    const float* __restrict__ ws_partial,
    const float* __restrict__ w_ca, const float* __restrict__ b_ca,
    float* __restrict__ out_ca)
{
    __shared__ float sPm[64];          // pooled mean [B][16]
    const int tid = threadIdx.x;
    const int blocksPerImage = HW / 128;   // 2048
    if (tid < 64) {
        int b = tid >> 4, c = tid & 15;
        float s = 0.f;
        for (int j = 0; j < blocksPerImage; ++j)
            s += ws_partial[((size_t)b * blocksPerImage + j) * 16 + c];
        sPm[tid] = s * (1.f / (float)HW);
    }
    __syncthreads();
    if (tid < BB * DIM) {
        int b = tid / DIM, d = tid % DIM;
        float a = b_ca[d];
        #pragma unroll
        for (int c = 0; c < 16; ++c) a += w_ca[d * 16 + c] * sPm[b * 16 + c];
        out_ca[tid] = sigmoidf(a);
    }
}

// ---------------------------------------------------------------------------
// Kernel 4: window MLP 64->8->2, softmax, +gumbel, hard argmax -> mask (class 0)
// ---------------------------------------------------------------------------
__global__ void __launch_bounds__(256) mask_kernel(
    const float* __restrict__ ws_xm,
    const float* __restrict__ gumbel,
    const float* __restrict__ w_m1, const float* __restrict__ b_m1,
    const float* __restrict__ w_m2, const float* __restrict__ b_m2,
    float* __restrict__ out_mask)
{
    const int idx = blockIdx.x * 256 + threadIdx.x;   // 0 .. B*NWIN-1
    const int b = idx / NWIN, n = idx % NWIN;
    const int wh = n >> 6, ww = n & 63;
    const float* base = ws_xm + (size_t)b * HW + (size_t)(wh * 8) * WID + ww * 8;

    float z[8];
    #pragma unroll
    for (int o = 0; o < 8; ++o) z[o] = b_m1[o];
    for (int dh = 0; dh < 8; ++dh) {
        const float* row = base + (size_t)dh * WID;
        #pragma unroll
        for (int dw = 0; dw < 8; ++dw) {
            float t = row[dw];
            int e = dh * 8 + dw;
            #pragma unroll
            for (int o = 0; o < 8; ++o) z[o] += w_m1[o * 64 + e] * t;
        }
    }
    #pragma unroll
    for (int o = 0; o < 8; ++o) z[o] = leaky(z[o]);

    float l0 = b_m2[0], l1 = b_m2[1];
    #pragma unroll
    for (int o = 0; o < 8; ++o) { l0 += w_m2[o] * z[o]; l1 += w_m2[8 + o] * z[o]; }

    const float mx = fmaxf(l0, l1);
    const float e0 = __expf(l0 - mx), e1 = __expf(l1 - mx);
    const float inv = 1.f / (e0 + e1);
    const float y0 = e0 * inv + gumbel[(size_t)idx * 2 + 0];
    const float y1 = e1 * inv + gumbel[(size_t)idx * 2 + 1];
    out_mask[idx] = (y0 >= y1) ? 1.0f : 0.0f;   // argmax ties -> class 0
}

// ---------------------------------------------------------------------------
extern "C" void kernel_launch(void* const* d_in, const int* in_sizes, int n_in,
                              void* d_out, int out_size, void* d_ws, size_t ws_size,
                              hipStream_t stream)
{
    const float* input_x = (const float*)d_in[0];
    const float* gumbel  = (const float*)d_in[1];
    const float* w_in    = (const float*)d_in[2];
    const float* b_in    = (const float*)d_in[3];
    const float* ln_w    = (const float*)d_in[4];
    const float* ln_b    = (const float*)d_in[5];
    const float* w_off1  = (const float*)d_in[6];
    const float* b_off1  = (const float*)d_in[7];
    const float* w_off2  = (const float*)d_in[8];
    const float* b_off2  = (const float*)d_in[9];
    const float* w_ca    = (const float*)d_in[10];
    const float* b_ca    = (const float*)d_in[11];
    const float* w_sa    = (const float*)d_in[12];
    const float* b_sa    = (const float*)d_in[13];
    const float* w_m1    = (const float*)d_in[14];
    const float* b_m1    = (const float*)d_in[15];
    const float* w_m2    = (const float*)d_in[16];
    const float* b_m2    = (const float*)d_in[17];

    // Output layout: mask[B*N], offsets[B*2*HW], ca[B*62], sa[B*HW]  (concatenated)
    float* out      = (float*)d_out;
    float* out_mask = out;
    float* out_off  = out_mask + (size_t)BB * NWIN;
    float* out_ca   = out_off + (size_t)BB * 2 * HW;
    float* out_sa   = out_ca + (size_t)BB * DIM;

    // Workspace layout (floats): partials[8192*16] | xm[B*HW] | xact[B*HW*16]
    float* ws         = (float*)d_ws;
    float* ws_partial = ws;
    float* ws_xm      = ws_partial + (size_t)BB * (HW / 128) * 16;
    float* ws_xact    = ws_xm + (size_t)BB * HW;

    main_kernel<<<BB * (HW / 128), 256, 0, stream>>>(
        input_x, w_in, b_in, ln_w, ln_b, w_off1, b_off1, w_off2, b_off2,
        out_off, ws_partial, ws_xm, ws_xact);

    sa_kernel<<<(BB * HW) / 256, 256, 0, stream>>>(ws_xact, w_sa, b_sa, out_sa);

    ca_kernel<<<1, 256, 0, stream>>>(ws_partial, w_ca, b_ca, out_ca);

    mask_kernel<<<(BB * NWIN) / 256, 256, 0, stream>>>(
        ws_xm, gumbel, w_m1, b_m1, w_m2, b_m2, out_mask);
}